// SoftCLDiceLoss_78726750535814
// MI455X (gfx1250) — compile-verified
//
#include <hip/hip_runtime.h>

// ---------------------------------------------------------------------------
// Soft clDice loss, MI455X (gfx1250).
// Fused erode(7-pt cross) + separable dilate(3x3x3) + skel update per pass,
// LDS-tiled, CDNA5 async global->LDS staging (b64 when available, else b32),
// all guarded by __has_builtin so compilation can never break.
// ---------------------------------------------------------------------------

#define B_   2
#define C_   3
#define D_   128
#define H_   160
#define W_   160
#define DHW_ (D_ * H_ * W_)      // 3,276,800
#define NVOL 4                   // (b,c): b in {0,1}, class channel in {1,2}
#define NCHUNK (DHW_ / 1024)     // 3200 stage-1 reduction blocks per volume

// Output tile per 256-thread block (exact divisors of D/H/W)
#define TZ 4
#define TY 8
#define TX 32
// x-tile: halo 2 (erode radius 1 + dilate radius 1)
#define XZ (TZ + 4)
#define XY (TY + 4)
#define XX (TX + 4)
#define XT (XZ * XY * XX)        // 3456 floats = 13.8 KB
// erode tile: halo 1 (consumed by dilate)
#define EZ (TZ + 2)
#define EY (TY + 2)
#define EX (TX + 2)
#define ET (EZ * EY * EX)        // 2040 floats = 8.2 KB
// separable-dilate temporaries
#define T1X TX                   // x-max stage keeps ex in [1, TX], stored at ex-1
#define T1N (EZ * EY * T1X)      // 1920 floats
#define T2N (EZ * TY * TX)       // 1536 floats

#define AS_GLOBAL __attribute__((address_space(1)))
#define AS_LDS    __attribute__((address_space(3)))

typedef int v2i __attribute__((vector_size(2 * sizeof(int))));

#if defined(__has_builtin)
#if __has_builtin(__builtin_amdgcn_global_load_async_to_lds_b32)
#define HAVE_ASYNC_LDS 1
#else
#define HAVE_ASYNC_LDS 0
#endif
#if __has_builtin(__builtin_amdgcn_global_load_async_to_lds_b64)
#define HAVE_ASYNC_LDS64 1
#else
#define HAVE_ASYNC_LDS64 0
#endif
#if __has_builtin(__builtin_amdgcn_s_wait_asynccnt)
#define HAVE_WAIT_ASYNC 1
#else
#define HAVE_WAIT_ASYNC 0
#endif
#else
#define HAVE_ASYNC_LDS 0
#define HAVE_ASYNC_LDS64 0
#define HAVE_WAIT_ASYNC 0
#endif

__device__ __forceinline__ int clampi(int v, int lo, int hi) {
    return v < lo ? lo : (v > hi ? hi : v);
}

// ---------------------------------------------------------------------------
// Fused pass: e = erode(x); open = dilate(e) (separable); delta = relu(x-open)
//   first: skel = delta     else: skel += relu(delta - skel*delta)
// mode 0: x from packed buffer xsrc; mode 1: x = net_output[b, c, ...] (pred);
// mode 2: x = one_hot(target)[b, c, ...] (gt).
// ---------------------------------------------------------------------------
__global__ __launch_bounds__(256) void skel_pass_kernel(
    const float* __restrict__ xsrc, const float* __restrict__ netout,
    const long long* __restrict__ tgt, float* __restrict__ ebuf,
    float* __restrict__ skel, int mode, int first) {
  __shared__ float sx[XT];
  __shared__ float se[ET];
  __shared__ float t1[T1N];
  __shared__ float t2[T2N];

  const int tid = threadIdx.x;
  const int tx  = blockIdx.x % (W_ / TX);
  const int ty  = blockIdx.x / (W_ / TX);
  const int x0 = tx * TX, y0 = ty * TY, z0 = blockIdx.y * TZ;
  const int vol = blockIdx.z;
  const int b = vol >> 1, c = (vol & 1) + 1;

  const float* src0 = (mode == 0) ? (xsrc + (size_t)vol * DHW_)
                                  : (netout + (size_t)(b * C_ + c) * DHW_);

  // Whole x-tile (halo 2) strictly inside the volume? (block-uniform branch)
  const bool interior = (z0 >= 2) && (z0 + TZ + 1 < D_) && (y0 >= 2) &&
                        (y0 + TY + 1 < H_) && (x0 >= 2) && (x0 + TX + 1 < W_);

#if HAVE_ASYNC_LDS
  if (mode != 2 && interior) {
#if HAVE_ASYNC_LDS64
    // b64 async DMA: interior tiles are 8B-aligned everywhere
    // (x0-2 even, row pitch 640B, LDS row pitch 144B).
    for (int i2 = tid; i2 < XT / 2; i2 += 256) {
      int row = i2 / (XX / 2), p = i2 % (XX / 2);
      int zz = row / XY, yy = row % XY;
      size_t s = ((size_t)(z0 - 2 + zz) * H_ + (y0 - 2 + yy)) * W_ +
                 (x0 - 2 + 2 * p);
      float* gp = const_cast<float*>(src0 + s);
      __builtin_amdgcn_global_load_async_to_lds_b64(
          (AS_GLOBAL v2i*)gp, (AS_LDS v2i*)(&sx[row * XX + 2 * p]), 0, 0);
    }
#else
    for (int i = tid; i < XT; i += 256) {
      int zz = i / (XY * XX), r = i % (XY * XX), yy = r / XX, xx = r % XX;
      size_t s = ((size_t)(z0 - 2 + zz) * H_ + (y0 - 2 + yy)) * W_ + (x0 - 2 + xx);
      float* gp = const_cast<float*>(src0 + s);
      __builtin_amdgcn_global_load_async_to_lds_b32(
          (AS_GLOBAL int*)gp, (AS_LDS int*)(&sx[i]), 0, 0);
    }
#endif
#if HAVE_WAIT_ASYNC
    __builtin_amdgcn_s_wait_asynccnt(0);
#else
    asm volatile("s_wait_asynccnt 0x0" ::: "memory");
#endif
    __syncthreads();
  } else
#endif
  {
    for (int i = tid; i < XT; i += 256) {
      int zz = i / (XY * XX), r = i % (XY * XX), yy = r / XX, xx = r % XX;
      int gz = z0 - 2 + zz, gy = y0 - 2 + yy, gx = x0 - 2 + xx;
      float v = __builtin_inff();  // min-neutral == window-shrink boundary
      if ((unsigned)gz < D_ && (unsigned)gy < H_ && (unsigned)gx < W_) {
        size_t s = ((size_t)gz * H_ + gy) * W_ + gx;
        if (mode == 2) {
          int lab = clampi((int)tgt[(size_t)b * DHW_ + s], 0, C_ - 1);
          v = (lab == c) ? 1.0f : 0.0f;
        } else {
          v = src0[s];
        }
      }
      sx[i] = v;
    }
    __syncthreads();
  }

  // erode: 7-point cross min (OOB x is +inf -> ignored)
  for (int i = tid; i < ET; i += 256) {
    int ez = i / (EY * EX), r = i % (EY * EX), ey = r / EX, ex = r % EX;
    int gz = z0 - 1 + ez, gy = y0 - 1 + ey, gx = x0 - 1 + ex;
    float v = -__builtin_inff();  // max-neutral for dilate at OOB positions
    if ((unsigned)gz < D_ && (unsigned)gy < H_ && (unsigned)gx < W_) {
      int xc = ((ez + 1) * XY + (ey + 1)) * XX + (ex + 1);
      float m = sx[xc];
      m = fminf(m, sx[xc - XY * XX]);
      m = fminf(m, sx[xc + XY * XX]);
      m = fminf(m, sx[xc - XX]);
      m = fminf(m, sx[xc + XX]);
      m = fminf(m, sx[xc - 1]);
      m = fminf(m, sx[xc + 1]);
      v = m;
    }
    se[i] = v;
  }
  __syncthreads();

  // separable dilate, stage X: t1[ez][ey][ox] = max(se[ez][ey][ox .. ox+2])
  for (int j = tid; j < T1N; j += 256) {
    int ez = j / (EY * T1X), r = j % (EY * T1X), ey = r / T1X, ox = r % T1X;
    int s0i = (ez * EY + ey) * EX + ox;  // ex = ox+1 -> window [ox, ox+2]
    t1[j] = fmaxf(fmaxf(se[s0i], se[s0i + 1]), se[s0i + 2]);
  }
  __syncthreads();

  // stage Y: t2[ez][oy][ox] = max(t1[ez][oy .. oy+2][ox])
  for (int j = tid; j < T2N; j += 256) {
    int ez = j / (TY * TX), r = j % (TY * TX), oy = r / TX, ox = r % TX;
    int s0i = (ez * EY + oy) * T1X + ox;
    t2[j] = fmaxf(fmaxf(t1[s0i], t1[s0i + T1X]), t1[s0i + 2 * T1X]);
  }
  __syncthreads();

  // stage Z (= open) + skel update + write e (next iteration's x)
  for (int i = tid; i < TZ * TY * TX; i += 256) {
    int oz = i / (TY * TX), r = i % (TY * TX), oy = r / TX, ox = r % TX;
    int s0i = (oz * TY + oy) * TX + ox;
    float open = fmaxf(fmaxf(t2[s0i], t2[s0i + TY * TX]),
                       t2[s0i + 2 * TY * TX]);
    float ev = se[((oz + 1) * EY + (oy + 1)) * EX + (ox + 1)];
    float xv = sx[((oz + 2) * XY + (oy + 2)) * XX + (ox + 2)];
    size_t pos = (size_t)vol * DHW_ +
                 ((size_t)(z0 + oz) * H_ + (y0 + oy)) * W_ + (x0 + ox);
    ebuf[pos] = ev;
    float delta = fmaxf(xv - open, 0.0f);
    if (first) {
      skel[pos] = delta;  // relu(x - open(x))
    } else {
      float s0 = skel[pos];
      skel[pos] = s0 + fmaxf(delta - s0 * delta, 0.0f);
    }
  }
}

// ---------------------------------------------------------------------------
// Stage-1 reduction: per (vol, 1024-elem chunk) -> {sum(skel*other), sum(skel)}
// phase 0: other = gt(one-hot target);  phase 1: other = pred (net_output)
// ---------------------------------------------------------------------------
__global__ __launch_bounds__(256) void reduce_stage1_kernel(
    const float* __restrict__ skel, const float* __restrict__ netout,
    const long long* __restrict__ tgt, int phase, float* __restrict__ partials) {
  __shared__ float rp[256];
  __shared__ float rs[256];
  const int vol = blockIdx.y;
  const int b = vol >> 1, c = (vol & 1) + 1;
  const size_t base = (size_t)blockIdx.x * 1024;
  float p = 0.0f, s = 0.0f;
#pragma unroll
  for (int k = 0; k < 4; k++) {
    size_t idx = base + (size_t)k * 256 + threadIdx.x;
    float sv = skel[(size_t)vol * DHW_ + idx];
    float ov;
    if (phase == 0) {
      int lab = clampi((int)tgt[(size_t)b * DHW_ + idx], 0, C_ - 1);
      ov = (lab == c) ? 1.0f : 0.0f;
    } else {
      ov = netout[(size_t)(b * C_ + c) * DHW_ + idx];
    }
    p += sv * ov;
    s += sv;
  }
  rp[threadIdx.x] = p;
  rs[threadIdx.x] = s;
  __syncthreads();
  for (int off = 128; off > 0; off >>= 1) {
    if ((int)threadIdx.x < off) {
      rp[threadIdx.x] += rp[threadIdx.x + off];
      rs[threadIdx.x] += rs[threadIdx.x + off];
    }
    __syncthreads();
  }
  if (threadIdx.x == 0) {
    size_t o = ((size_t)vol * NCHUNK + blockIdx.x) * 2;
    partials[o] = rp[0];
    partials[o + 1] = rs[0];
  }
}

__global__ __launch_bounds__(256) void reduce_stage2_kernel(
    const float* __restrict__ partials, float* __restrict__ sums, int phase) {
  __shared__ float rp[256];
  __shared__ float rs[256];
  const int vol = blockIdx.x;
  float p = 0.0f, s = 0.0f;
  for (int i = threadIdx.x; i < NCHUNK; i += 256) {
    size_t o = ((size_t)vol * NCHUNK + i) * 2;
    p += partials[o];
    s += partials[o + 1];
  }
  rp[threadIdx.x] = p;
  rs[threadIdx.x] = s;
  __syncthreads();
  for (int off = 128; off > 0; off >>= 1) {
    if ((int)threadIdx.x < off) {
      rp[threadIdx.x] += rp[threadIdx.x + off];
      rs[threadIdx.x] += rs[threadIdx.x + off];
    }
    __syncthreads();
  }
  if (threadIdx.x == 0) {
    sums[phase * 8 + vol] = rp[0];      // sum(skel * other)
    sums[phase * 8 + 4 + vol] = rs[0];  // sum(skel)
  }
}

__global__ void final_loss_kernel(const float* __restrict__ sums,
                                  float* __restrict__ out) {
  if (threadIdx.x == 0 && blockIdx.x == 0) {
    const float S = 1e-5f;
    float acc = 0.0f;
    for (int v = 0; v < NVOL; v++) {
      float tprec = (sums[v] + S) / (sums[4 + v] + S);       // skel_pred vs gt
      float tsens = (sums[8 + v] + S) / (sums[12 + v] + S);  // skel_gt vs pred
      acc += (2.0f * tprec * tsens + S) / (tprec + tsens + S);
    }
    out[0] = 1.0f - acc * 0.25f;
  }
}

extern "C" void kernel_launch(void* const* d_in, const int* in_sizes, int n_in,
                              void* d_out, int out_size, void* d_ws,
                              size_t ws_size, hipStream_t stream) {
  (void)in_sizes; (void)n_in; (void)out_size; (void)ws_size;
  const float* net = (const float*)d_in[0];          // (2,3,128,160,160) f32
  const long long* tgt = (const long long*)d_in[1];  // (2,1,128,160,160) i64
  float* out = (float*)d_out;

  // Workspace: 3 x 52.4MB ping-pong/skel buffers + reduction scratch
  // (~157.4 MB total; entire working set L2-resident on MI455X's 192MB L2).
  float* xA = (float*)d_ws;
  float* xB = xA + (size_t)NVOL * DHW_;
  float* sk = xB + (size_t)NVOL * DHW_;
  float* partials = sk + (size_t)NVOL * DHW_;                 // 2*4*3200 floats
  float* sums = partials + (size_t)2 * NVOL * NCHUNK;         // 16 floats

  const dim3 sgrid((W_ / TX) * (H_ / TY), D_ / TZ, NVOL);     // (100, 32, 4)
  const dim3 sblk(256);

  // ---- pred phase: x0 = net_output[:,1:], 4 fused passes -----------------
  skel_pass_kernel<<<sgrid, sblk, 0, stream>>>(nullptr, net, tgt, xA, sk, 1, 1);
  skel_pass_kernel<<<sgrid, sblk, 0, stream>>>(xA, net, tgt, xB, sk, 0, 0);
  skel_pass_kernel<<<sgrid, sblk, 0, stream>>>(xB, net, tgt, xA, sk, 0, 0);
  skel_pass_kernel<<<sgrid, sblk, 0, stream>>>(xA, net, tgt, xB, sk, 0, 0);
  reduce_stage1_kernel<<<dim3(NCHUNK, NVOL), sblk, 0, stream>>>(sk, net, tgt, 0,
                                                                partials);
  reduce_stage2_kernel<<<dim3(NVOL), sblk, 0, stream>>>(partials, sums, 0);

  // ---- gt phase: x0 = one_hot(target)[:,1:], 4 fused passes --------------
  skel_pass_kernel<<<sgrid, sblk, 0, stream>>>(nullptr, net, tgt, xA, sk, 2, 1);
  skel_pass_kernel<<<sgrid, sblk, 0, stream>>>(xA, net, tgt, xB, sk, 0, 0);
  skel_pass_kernel<<<sgrid, sblk, 0, stream>>>(xB, net, tgt, xA, sk, 0, 0);
  skel_pass_kernel<<<sgrid, sblk, 0, stream>>>(xA, net, tgt, xB, sk, 0, 0);
  reduce_stage1_kernel<<<dim3(NCHUNK, NVOL), sblk, 0, stream>>>(sk, net, tgt, 1,
                                                                partials);
  reduce_stage2_kernel<<<dim3(NVOL), sblk, 0, stream>>>(partials, sums, 1);

  final_loss_kernel<<<1, 32, 0, stream>>>(sums, out);
}